// LaplacianRegularizationLoss_69080253989044
// MI455X (gfx1250) — compile-verified
//
#include <hip/hip_runtime.h>
#include <math.h>

#define N_PTS 8192
#define DIM   128
#define KNN   10
#define BR    64      // rows per workgroup (= threads per workgroup, 2 waves)
#define CP    64      // columns per streamed panel
#define LDA   132     // padded LDS row stride (floats) for A/B tiles
#define LDD   65      // padded LDS row stride for distance tile
#define NPANEL (N_PTS / CP)

typedef __attribute__((ext_vector_type(2))) float v2f;
typedef __attribute__((ext_vector_type(8))) float v8f;

// ---------------- Kernel 0: row squared norms + zero the output ----------------
__global__ __launch_bounds__(256) void sq_init_kernel(const float* __restrict__ E,
                                                      float* __restrict__ sq,
                                                      float* __restrict__ out) {
  int i = blockIdx.x * blockDim.x + threadIdx.x;
  if (i == 0) out[0] = 0.0f;
  if (i < N_PTS) {
    const float4* row = (const float4*)(E + (size_t)i * DIM);
    float s = 0.0f;
#pragma unroll
    for (int c = 0; c < DIM / 4; ++c) {
      float4 v = row[c];
      s += v.x * v.x + v.y * v.y + v.z * v.z + v.w * v.w;
    }
    sq[i] = s;
  }
}

// Issue an async global->LDS copy of one CP x DIM panel (16B per lane per issue).
// GLOBAL_LOAD_ASYNC_TO_LDS_B128, tracked with ASYNCcnt; LDS written directly.
__device__ __forceinline__ void async_load_panel(const float* __restrict__ E,
                                                 int colbase, float* dstBase, int tid) {
  for (int v = tid; v < CP * (DIM / 4); v += BR) {
    int r = v >> 5;                    // DIM/4 == 32 chunks per row
    int c = (v & 31) << 2;             // float index within the row
    unsigned lds = (unsigned)(uintptr_t)(dstBase + r * LDA + c);
    unsigned long long gaddr =
        (unsigned long long)(uintptr_t)(E + (size_t)(colbase + r) * DIM + c);
    asm volatile("global_load_async_to_lds_b128 %0, %1, off"
                 :: "v"(lds), "v"(gaddr) : "memory");
  }
}

// ---------------- Kernel 1: fused WMMA Gram tiles -> d^2 -> streaming top-K ----------------
__global__ __launch_bounds__(BR) void knn_kernel(const float* __restrict__ E,
                                                 const float* __restrict__ sq,
                                                 int* __restrict__ knn_idx,
                                                 float* __restrict__ knn_d2) {
  __shared__ __align__(16) float lA[BR * LDA];
  __shared__ __align__(16) float lB[2 * CP * LDA];   // double-buffered async panel
  __shared__ float lD[BR * LDD];
  __shared__ float lSqRow[BR];

  const int tid  = threadIdx.x;
  const int lane = tid & 31;           // wave32
  const int wv   = tid >> 5;           // wave id: 0..1
  const int lh   = lane & 15;
  const int koff = (lane >> 4) << 1;   // 0 or 2 : K-pair selected by lane half
  const int rowbase = blockIdx.x * BR;

  // Kick off the async copy of panel 0 before touching anything else.
  async_load_panel(E, 0, lB, tid);

  // Cooperative load of the A row-block (BR x DIM) into LDS, float4-coalesced.
  for (int v = tid; v < BR * (DIM / 4); v += BR) {
    int r = v >> 5;                    // DIM/4 == 32
    int c = (v & 31) << 2;
    *(float4*)&lA[r * LDA + c] = *(const float4*)&E[(size_t)(rowbase + r) * DIM + c];
  }
  lSqRow[tid] = sq[rowbase + tid];

  // Register-resident sorted top-K (ascending d^2) for the row this thread owns.
  float bd[KNN];
  int   bj[KNN];
#pragma unroll
  for (int q = 0; q < KNN; ++q) { bd[q] = 3.0e38f; bj[q] = 0; }

  for (int p = 0; p < NPANEL; ++p) {
    const int colbase = p * CP;
    const float* lBcur = lB + (p & 1) * (CP * LDA);

    // Drain the in-flight panel copy, then make it (and lA/lD phases) visible.
    asm volatile("s_wait_asynccnt 0x0" ::: "memory");
    __syncthreads();

    // Overlap the next panel's copy with this panel's WMMA + top-K scan.
    if (p + 1 < NPANEL)
      async_load_panel(E, colbase + CP, lB + ((p + 1) & 1) * (CP * LDA), tid);

    // 4x4 grid of 16x16 output tiles; 8 tiles per wave.
    for (int wt = wv * 8; wt < wv * 8 + 8; ++wt) {
      const int rt = wt >> 2;
      const int ct = wt & 3;
      const float* Ab = &lA[(rt * 16 + lh) * LDA + koff];
      const float* Bb = &lBcur[(ct * 16 + lh) * LDA + koff];
      v8f acc = {0.f, 0.f, 0.f, 0.f, 0.f, 0.f, 0.f, 0.f};
#pragma unroll
      for (int k0 = 0; k0 < DIM; k0 += 4) {
        v2f a = *(const v2f*)(Ab + k0);
        v2f b = *(const v2f*)(Bb + k0);
        acc = __builtin_amdgcn_wmma_f32_16x16x4_f32(false, a, false, b,
                                                    (short)0, acc, false, false);
      }
      // C/D layout: vgpr r, lane L -> M = rt*16 + r + 8*(L>=16), N = ct*16 + (L&15)
      const int cia  = ct * 16 + lh;
      const int gcol = colbase + cia;
      const float sqc = sq[gcol];                 // L2-resident
      const int rb2 = rt * 16 + ((lane >> 4) << 3);
#pragma unroll
      for (int r = 0; r < 8; ++r) {
        const int rib = rb2 + r;
        float d2v = lSqRow[rib] + sqc - 2.0f * acc[r];
        d2v = fmaxf(d2v, 0.0f);
        if (rowbase + rib == gcol) d2v = 3.0e38f; // exclude self
        lD[rib * LDD + cia] = d2v;
      }
    }
    __syncthreads();

    // Each thread scans its own row's CP candidates and updates its top-K.
    // (Next iteration's loop-top barrier separates this from lD being rewritten.)
    for (int c = 0; c < CP; ++c) {
      float cd = lD[tid * LDD + c];
      if (cd < bd[KNN - 1]) {
        bd[KNN - 1] = cd;
        bj[KNN - 1] = colbase + c;
#pragma unroll
        for (int q = KNN - 1; q > 0; --q) {
          if (bd[q] < bd[q - 1]) {
            float td = bd[q]; bd[q] = bd[q - 1]; bd[q - 1] = td;
            int   tj = bj[q]; bj[q] = bj[q - 1]; bj[q - 1] = tj;
          }
        }
      }
    }
  }

  const int row = rowbase + tid;
#pragma unroll
  for (int q = 0; q < KNN; ++q) {
    knn_idx[row * KNN + q] = bj[q];
    knn_d2[row * KNN + q]  = bd[q];
  }
}

// ---------------- Kernel 2: symmetrized edge sum ----------------
// loss = 0.5 * sum_ij A_ij d2_ij with A = max(W, W^T); weights are symmetric in (i,j),
// so per directed kNN edge: mutual pairs contribute 0.5*w*d2 (counted twice),
// one-directional pairs contribute w*d2 once.
__global__ __launch_bounds__(256) void edge_loss_kernel(const int* __restrict__ knn_idx,
                                                        const float* __restrict__ knn_d2,
                                                        float* __restrict__ out) {
  __shared__ float red[256];
  int t = blockIdx.x * blockDim.x + threadIdx.x;
  float contrib = 0.0f;
  if (t < N_PTS * KNN) {
    int i = t / KNN;
    int j = knn_idx[t];
    float d2 = knn_d2[t];
    float w = __expf(-d2 * 0.5f);      // SIGMA = 1
    bool mutual = false;
#pragma unroll
    for (int q = 0; q < KNN; ++q) mutual = mutual || (knn_idx[j * KNN + q] == i);
    contrib = (mutual ? 0.5f : 1.0f) * w * d2;
  }
  red[threadIdx.x] = contrib;
  __syncthreads();
  for (int s = 128; s > 0; s >>= 1) {
    if (threadIdx.x < s) red[threadIdx.x] += red[threadIdx.x + s];
    __syncthreads();
  }
  if (threadIdx.x == 0)
    atomicAdd(out, red[0] * (1.0f / ((float)N_PTS * (float)N_PTS)));
}

// ---------------- Launch ----------------
extern "C" void kernel_launch(void* const* d_in, const int* in_sizes, int n_in,
                              void* d_out, int out_size, void* d_ws, size_t ws_size,
                              hipStream_t stream) {
  (void)in_sizes; (void)n_in; (void)out_size; (void)ws_size;
  const float* E = (const float*)d_in[0];
  float* out = (float*)d_out;

  float* d_sq    = (float*)d_ws;                                               // N floats
  int*   knn_idx = (int*)((char*)d_ws + (size_t)N_PTS * 4);                    // N*K ints
  float* knn_d2  = (float*)((char*)d_ws + (size_t)N_PTS * 4
                                        + (size_t)N_PTS * KNN * 4);            // N*K floats

  sq_init_kernel<<<(N_PTS + 255) / 256, 256, 0, stream>>>(E, d_sq, out);
  knn_kernel<<<N_PTS / BR, BR, 0, stream>>>(E, d_sq, knn_idx, knn_d2);
  edge_loss_kernel<<<(N_PTS * KNN + 255) / 256, 256, 0, stream>>>(knn_idx, knn_d2, out);
}